// GraphConv_78486232367383
// MI455X (gfx1250) — compile-verified
//
#include <hip/hip_runtime.h>
#include <hip/hip_bf16.h>
#include <stdint.h>

typedef __attribute__((ext_vector_type(2))) float v2f;
typedef __attribute__((ext_vector_type(8))) float v8f;

#define D 64
#define WAVES_PER_BLOCK 8

// Native device-scope fp32 atomic add (avoid any CAS-loop expansion).
__device__ __forceinline__ void atomic_add_f32_dev(float* p, float v) {
    asm volatile("global_atomic_add_f32 %0, %1, off scope:SCOPE_DEV"
                 :: "v"(p), "v"(v) : "memory");
}

// Kernel 1: fused  out_w0 = verts@W0 + b0   (-> d_out)
//                  out_w1 = verts@W1 + b1   (-> workspace)
// One wave computes a 16(M) x 64(N) strip for BOTH weight matrices using
// V_WMMA_F32_16X16X4_F32, accumulating over K=64 in 16 steps.
__global__ __launch_bounds__(256)
void gc_gemm_kernel(const float* __restrict__ verts,
                    const float* __restrict__ w0k, const float* __restrict__ w0b,
                    const float* __restrict__ w1k, const float* __restrict__ w1b,
                    float* __restrict__ outw0,
                    float* __restrict__ outw1,
                    int nverts)
{
    __shared__ float ldsW[2 * D * D];   // 32 KB: W0 then W1, row-major [K][N]
    {
        const float4* s0 = (const float4*)w0k;
        const float4* s1 = (const float4*)w1k;
        float4* t0 = (float4*)ldsW;
        float4* t1 = (float4*)(ldsW + D * D);
        for (int i = threadIdx.x; i < (D * D) / 4; i += 256) {
            t0[i] = s0[i];
            t1[i] = s1[i];
        }
    }
    __syncthreads();

    const int lane  = threadIdx.x & 31;
    const int wave  = threadIdx.x >> 5;
    const int strip = blockIdx.x * WAVES_PER_BLOCK + wave;   // 16-row strip id
    if (strip * 16 >= nverts) return;   // wave-uniform: EXEC stays all-ones

    const int half = lane >> 4;         // 0: lanes 0-15, 1: lanes 16-31
    const int l16  = lane & 15;
    const int arow = strip * 16 + l16;  // A-matrix row owned by this lane

    // Accumulators: 4 N-tiles x (16x16 f32) per weight matrix; init with bias.
    v8f acc0[4], acc1[4];
    #pragma unroll
    for (int nt = 0; nt < 4; ++nt) {
        const float bv0 = w0b[nt * 16 + l16];
        const float bv1 = w1b[nt * 16 + l16];
        #pragma unroll
        for (int r = 0; r < 8; ++r) { acc0[nt][r] = bv0; acc1[nt][r] = bv1; }
    }

    const float* apv = verts + (long long)arow * D;
    const float* lw0 = ldsW;
    const float* lw1 = ldsW + D * D;

    for (int k0 = 0; k0 < D; k0 += 4) {
        const int ka = k0 + 2 * half;                 // this lane's K base
        const v2f afrag = *(const v2f*)(apv + ka);    // A: {K=ka, K=ka+1}
        #pragma unroll
        for (int nt = 0; nt < 4; ++nt) {
            const int n = nt * 16 + l16;              // B/C column for this lane
            v2f b0f, b1f;
            b0f[0] = lw0[(ka + 0) * D + n];
            b0f[1] = lw0[(ka + 1) * D + n];
            b1f[0] = lw1[(ka + 0) * D + n];
            b1f[1] = lw1[(ka + 1) * D + n];
            acc0[nt] = __builtin_amdgcn_wmma_f32_16x16x4_f32(
                false, afrag, false, b0f, (short)0, acc0[nt], false, false);
            acc1[nt] = __builtin_amdgcn_wmma_f32_16x16x4_f32(
                false, afrag, false, b1f, (short)0, acc1[nt], false, false);
        }
    }

    // C/D layout: component r -> row (r + 8*half), column l16 within tile.
    #pragma unroll
    for (int nt = 0; nt < 4; ++nt) {
        const int n = nt * 16 + l16;
        #pragma unroll
        for (int r = 0; r < 8; ++r) {
            const long long m = (long long)(strip * 16 + r + 8 * half);
            outw0[m * D + n] = acc0[nt][r];
            outw1[m * D + n] = acc1[nt][r];
        }
    }
}

// Kernel 2: symmetric neighbor scatter-add.
// 16 threads per edge, 4 features each: out[s] += vw1[d], out[d] += vw1[s].
__global__ __launch_bounds__(256)
void gc_edge_kernel(const int* __restrict__ edges,
                    const float* __restrict__ vw1,
                    float* __restrict__ out,
                    int nedges)
{
    const unsigned t = blockIdx.x * 256u + threadIdx.x;
    const int e = (int)(t >> 4);
    if (e >= nedges) return;
    const int f = (int)(t & 15u) * 4;

    const int s = edges[2 * e + 0];
    const int d = edges[2 * e + 1];

    const float4 vd = *(const float4*)(vw1 + (long long)d * D + f);
    const float4 vs = *(const float4*)(vw1 + (long long)s * D + f);
    float* ps = out + (long long)s * D + f;
    float* pd = out + (long long)d * D + f;

    atomic_add_f32_dev(ps + 0, vd.x);
    atomic_add_f32_dev(ps + 1, vd.y);
    atomic_add_f32_dev(ps + 2, vd.z);
    atomic_add_f32_dev(ps + 3, vd.w);
    atomic_add_f32_dev(pd + 0, vs.x);
    atomic_add_f32_dev(pd + 1, vs.y);
    atomic_add_f32_dev(pd + 2, vs.z);
    atomic_add_f32_dev(pd + 3, vs.w);
}

extern "C" void kernel_launch(void* const* d_in, const int* in_sizes, int n_in,
                              void* d_out, int out_size, void* d_ws, size_t ws_size,
                              hipStream_t stream) {
    const float* verts = (const float*)d_in[0];
    const int*   edges = (const int*)d_in[1];
    const float* w0k   = (const float*)d_in[2];
    const float* w0b   = (const float*)d_in[3];
    const float* w1k   = (const float*)d_in[4];
    const float* w1b   = (const float*)d_in[5];
    float* out = (float*)d_out;
    float* vw1 = (float*)d_ws;          // V*64 floats = 25.6 MB scratch

    const int nverts = in_sizes[0] / D;     // 100000
    const int nedges = in_sizes[1] / 2;     // 1000000

    const int strips  = (nverts + 15) / 16;
    const int blocks1 = (strips + WAVES_PER_BLOCK - 1) / WAVES_PER_BLOCK;
    gc_gemm_kernel<<<blocks1, 256, 0, stream>>>(verts, w0k, w0b, w1k, w1b,
                                                out, vw1, nverts);

    const long long threads2 = (long long)nedges * 16;
    const int blocks2 = (int)((threads2 + 255) / 256);
    gc_edge_kernel<<<blocks2, 256, 0, stream>>>(edges, vw1, out, nedges);
}